// LiftSplat_39771397161466
// MI455X (gfx1250) — compile-verified
//
#include <hip/hip_runtime.h>

typedef __attribute__((ext_vector_type(2))) float v2f;
typedef __attribute__((ext_vector_type(8))) float v8f;

#define NBN          24          // B*N
#define PTS_PER_CAM  28864       // 41*16*44
#define WAVES_PER_CAM 1804       // 28864/16
#define NPTS         (NBN * PTS_PER_CAM)   // 692736
#define C_CH         64
#define OUT_PER_B    (C_CH * 200 * 200)    // 2,560,000

// ---------------------------------------------------------------- zero output
__global__ void lss_zero(float4* __restrict__ out) {
    size_t i = (size_t)blockIdx.x * blockDim.x + threadIdx.x;
    out[i] = make_float4(0.f, 0.f, 0.f, 0.f);
}

// ------------------------------------------------- per-(b,n) affine transforms
__device__ inline void inv3x3(const float* m, float* inv) {
    float c00 = m[4]*m[8] - m[5]*m[7];
    float c01 = m[5]*m[6] - m[3]*m[8];
    float c02 = m[3]*m[7] - m[4]*m[6];
    float id  = 1.0f / (m[0]*c00 + m[1]*c01 + m[2]*c02);
    inv[0] = c00*id; inv[1] = (m[2]*m[7]-m[1]*m[8])*id; inv[2] = (m[1]*m[5]-m[2]*m[4])*id;
    inv[3] = c01*id; inv[4] = (m[0]*m[8]-m[2]*m[6])*id; inv[5] = (m[2]*m[3]-m[0]*m[5])*id;
    inv[6] = c02*id; inv[7] = (m[1]*m[6]-m[0]*m[7])*id; inv[8] = (m[0]*m[4]-m[1]*m[3])*id;
}

__global__ void lss_setup(const float* __restrict__ rots, const float* __restrict__ trans,
                          const float* __restrict__ intrins, const float* __restrict__ post_rots,
                          const float* __restrict__ post_trans, float* __restrict__ aff) {
    int bn = threadIdx.x;
    if (bn >= NBN) return;
    float P[9], Pinv[9], K[9], Kinv[9], R[9], Cm[9];
    for (int i = 0; i < 9; ++i) {
        P[i] = post_rots[bn*9 + i];
        K[i] = intrins[bn*9 + i];
        R[i] = rots[bn*9 + i];
    }
    inv3x3(P, Pinv);
    inv3x3(K, Kinv);
    for (int r = 0; r < 3; ++r)
        for (int c = 0; c < 3; ++c)
            Cm[r*3+c] = R[r*3+0]*Kinv[0*3+c] + R[r*3+1]*Kinv[1*3+c] + R[r*3+2]*Kinv[2*3+c];
    float pt0 = post_trans[bn*3+0], pt1 = post_trans[bn*3+1], pt2 = post_trans[bn*3+2];
    float* A1 = aff + bn*24;   // stage 1: p' = Pinv*(f - post_trans)
    float* A2 = A1 + 12;       // stage 2: ego = (R*Kinv)*q + trans
    for (int r = 0; r < 3; ++r) {
        A1[r*4+0] = Pinv[r*3+0]; A1[r*4+1] = Pinv[r*3+1]; A1[r*4+2] = Pinv[r*3+2];
        A1[r*4+3] = -(Pinv[r*3+0]*pt0 + Pinv[r*3+1]*pt1 + Pinv[r*3+2]*pt2);
        A2[r*4+0] = Cm[r*3+0];   A2[r*4+1] = Cm[r*3+1];   A2[r*4+2] = Cm[r*3+2];
        A2[r*4+3] = trans[bn*3+r];
    }
}

// --------------------------------------------- geometry: WMMA affine pipeline
// One wave -> 16 points of one (b,n) camera. Two V_WMMA_F32_16X16X4_F32:
//   D1 = A1(3x4 affine, zero-padded to 16x4) x B1(4x16 homogeneous frustum pts)
//   unproject elementwise:  q = (x*z, y*z, z, 1)
//   D2 = A2 x B2  ->  ego coords, one point per lane (lanes 0-15)
__global__ void __launch_bounds__(256) lss_geom(const float* __restrict__ aff,
                                                int* __restrict__ idx) {
    const int lane  = threadIdx.x & 31;
    const int gwave = blockIdx.x * (blockDim.x >> 5) + (threadIdx.x >> 5);
    const int bn = gwave / WAVES_PER_CAM;
    const int lw = gwave - bn * WAVES_PER_CAM;

    const int n  = lane & 15;     // point column / A row
    const int h  = lane >> 4;     // K-half select (K = 2*h + vgpr)
    const int li = lw * 16 + n;   // local point index within (b,n)

    // frustum decomposition: li = d*704 + h*44 + w
    const int d_i = li / 704;
    const int rem = li - d_i * 704;
    const int hh  = rem / 44;
    const int ww  = rem - hh * 44;
    const float fx = (float)((double)ww * (703.0 / 43.0)); // linspace(0,703,44)
    const float fy = (float)(hh * 17);                     // linspace(0,255,16)
    const float fd = 4.0f + (float)d_i;                    // arange(4,45)

    // A operands: A(m,k) -> vgpr k&1, lane (k>>1)*16 + m ; rows >=3 are zero
    const float* A1p = aff + bn * 24;
    const float* A2p = A1p + 12;
    v2f a1, a2;
    a1[0] = (n < 3) ? A1p[n*4 + (2*h + 0)] : 0.0f;
    a1[1] = (n < 3) ? A1p[n*4 + (2*h + 1)] : 0.0f;
    a2[0] = (n < 3) ? A2p[n*4 + (2*h + 0)] : 0.0f;
    a2[1] = (n < 3) ? A2p[n*4 + (2*h + 1)] : 0.0f;

    // B1: rows (x, y, depth, 1) of point n; B(k,n) -> vgpr k&1, lane (k>>1)*16+n
    v2f b1;
    b1[0] = (h == 0) ? fx : fd;
    b1[1] = (h == 0) ? fy : 1.0f;

    v8f c0 = {};
    v8f d1 = __builtin_amdgcn_wmma_f32_16x16x4_f32(false, a1, false, b1,
                                                   (short)0, c0, false, false);
    // unproject: lanes<16 hold p' of point n in d1[0..2]
    float z_src = __shfl(d1[2], lane & 15, 32);   // lanes 16-31 fetch p'z of point n
    v2f b2;
    b2[0] = (h == 0) ? d1[0] * d1[2] : z_src;     // rows 0 / 2: x*z, z
    b2[1] = (h == 0) ? d1[1] * d1[2] : 1.0f;      // rows 1 / 3: y*z, 1

    v8f d2 = __builtin_amdgcn_wmma_f32_16x16x4_f32(false, a2, false, b2,
                                                   (short)0, c0, false, false);

    if (h == 0) {   // lanes 0-15: ego coords of point n in d2[0..2]
        // trunc-toward-zero cast matches .astype(int32)
        int xi = (int)((d2[0] + 50.0f) / 0.5f);
        int yi = (int)((d2[1] + 50.0f) / 0.5f);
        int zi = (int)((d2[2] + 10.0f) / 20.0f);
        bool kept = (xi >= 0) & (xi < 200) & (yi >= 0) & (yi < 200) & (zi >= 0) & (zi < 1);
        int b = bn / 6;
        int base = b * OUT_PER_B + xi * 200 + yi;
        idx[bn * PTS_PER_CAM + li] = kept ? base : -1;
    }
}

// ------------------------------------------- scatter: coalesced read + L2 atomics
// 64 threads per point (one per channel) -> each wave is uniform per point.
__global__ void __launch_bounds__(256) lss_scatter(const float* __restrict__ feats,
                                                   const int* __restrict__ idx,
                                                   float* __restrict__ out) {
    const size_t t = (size_t)blockIdx.x * 256 + threadIdx.x;
    const size_t p = t >> 6;
    const int    c = (int)(t & 63);
    const int base = idx[p];
    if (base < 0) return;                      // wave-uniform branch
    const float v = feats[p * 64 + c];         // fully coalesced
    unsafeAtomicAdd(out + (size_t)base + (size_t)c * 40000, v);  // global_atomic_add_f32 in L2
}

// ---------------------------------------------------------------------- launch
extern "C" void kernel_launch(void* const* d_in, const int* in_sizes, int n_in,
                              void* d_out, int out_size, void* d_ws, size_t ws_size,
                              hipStream_t stream) {
    const float* cam_feats  = (const float*)d_in[0];
    const float* rots       = (const float*)d_in[1];
    const float* trans      = (const float*)d_in[2];
    const float* intrins    = (const float*)d_in[3];
    const float* post_rots  = (const float*)d_in[4];
    const float* post_trans = (const float*)d_in[5];
    float* out = (float*)d_out;

    float* aff = (float*)d_ws;                       // 24 * 24 floats
    int*   idx = (int*)((char*)d_ws + 4096);         // NPTS ints (~2.77 MB)

    // out_size = 4*64*200*200 = 10,240,000 floats = 2,560,000 float4
    lss_zero<<<10000, 256, 0, stream>>>((float4*)out);
    lss_setup<<<1, 32, 0, stream>>>(rots, trans, intrins, post_rots, post_trans, aff);
    // 24 cams * 1804 waves = 43,296 waves; 8 waves/block -> 5412 blocks
    lss_geom<<<5412, 256, 0, stream>>>(aff, idx);
    // 692,736 points * 64 ch / 256 threads = 173,184 blocks
    lss_scatter<<<173184, 256, 0, stream>>>(cam_feats, idx, out);
}